// HeteroDictLinear_72456098284154
// MI455X (gfx1250) — compile-verified
//
#include <hip/hip_runtime.h>

typedef __attribute__((ext_vector_type(2))) float v2f;
typedef __attribute__((ext_vector_type(8))) float v8f;

#define BM  128          // rows per workgroup
#define BN   64          // output channels per workgroup
#define KC   32          // K-chunk staged in LDS
#define LDT (KC + 2)     // padded LDS row pitch (floats)

// y[m, o] = sum_k x[m, k] * W[o, k] + b[o]
// WMMA mapping: A(16x4) = x-rows, B(4x16) = W-rows (i.e. W.T columns), D(16x16)=y tile.
__global__ __launch_bounds__(256, 2)
void hdl_gemm_f32(const float* __restrict__ X,
                  const float* __restrict__ Wt,     // (512, C) row-major
                  const float* __restrict__ Bias,   // (512,)
                  float* __restrict__ Y,            // (nrows, 512)
                  int nrows, int C)
{
    __shared__ float sx[BM * LDT];
    __shared__ float sw[BN * LDT];

    const int tid   = threadIdx.x;
    const int lane  = tid & 31;
    const int wave  = tid >> 5;      // 0..7
    const int wm    = wave & 3;      // M quadrant (32 rows each)
    const int wn    = wave >> 2;     // N half (32 cols each)
    const int lmod  = lane & 15;
    const int lhalf = lane >> 4;     // 0 or 1

    const int block_m = blockIdx.x * BM;
    const int block_n = blockIdx.y * BN;

    v8f acc00 = {}, acc01 = {}, acc10 = {}, acc11 = {};

    // staging coordinates: 8 threads cover one 32-float (128B) row chunk
    const int col  = (tid & 7) << 2;   // 0,4,...,28
    const int row0 = tid >> 3;         // 0..31

    for (int kc = 0; kc < C; kc += KC) {
        // ---- stage X tile: BM x KC (zero-fill out-of-range rows) ----
        #pragma unroll
        for (int it = 0; it < 4; ++it) {
            const int r  = row0 + it * 32;
            const int gr = block_m + r;
            float4 v = make_float4(0.f, 0.f, 0.f, 0.f);
            if (gr < nrows)
                v = *(const float4*)(X + (size_t)gr * C + (kc + col));
            float* p = &sx[r * LDT + col];
            p[0] = v.x; p[1] = v.y; p[2] = v.z; p[3] = v.w;
        }
        // ---- stage W tile: BN x KC (always in range; OUT=512 divisible by BN) ----
        #pragma unroll
        for (int it = 0; it < 2; ++it) {
            const int r = row0 + it * 32;
            const float4 v = *(const float4*)(Wt + (size_t)(block_n + r) * C + (kc + col));
            float* p = &sw[r * LDT + col];
            p[0] = v.x; p[1] = v.y; p[2] = v.z; p[3] = v.w;
        }
        __syncthreads();

        // ---- 8 k-steps of V_WMMA_F32_16X16X4_F32, 4 tiles per wave ----
        #pragma unroll
        for (int k = 0; k < KC; k += 4) {
            const int kk = k + (lhalf << 1);   // lanes 0-15: K=k,k+1 ; lanes 16-31: K=k+2,k+3
            const v2f a0 = *(const v2f*)&sx[(wm * 32 +      lmod) * LDT + kk];
            const v2f a1 = *(const v2f*)&sx[(wm * 32 + 16 + lmod) * LDT + kk];
            const v2f b0 = *(const v2f*)&sw[(wn * 32 +      lmod) * LDT + kk];
            const v2f b1 = *(const v2f*)&sw[(wn * 32 + 16 + lmod) * LDT + kk];
            acc00 = __builtin_amdgcn_wmma_f32_16x16x4_f32(false, a0, false, b0,
                        (short)0, acc00, false, false);
            acc01 = __builtin_amdgcn_wmma_f32_16x16x4_f32(false, a0, false, b1,
                        (short)0, acc01, false, false);
            acc10 = __builtin_amdgcn_wmma_f32_16x16x4_f32(false, a1, false, b0,
                        (short)0, acc10, false, false);
            acc11 = __builtin_amdgcn_wmma_f32_16x16x4_f32(false, a1, false, b1,
                        (short)0, acc11, false, false);
        }
        __syncthreads();
    }

    // ---- epilogue: bias + guarded store ----
    // D layout: VGPR r, lanes 0-15 -> M=r, N=lane ; lanes 16-31 -> M=r+8, N=lane-16
    auto store_tile = [&](const v8f& a, int mtile, int ntile) {
        const int ocol  = block_n + wn * 32 + ntile * 16 + lmod;
        const float bv  = Bias[ocol];
        const int mbase = block_m + wm * 32 + mtile * 16 + (lhalf << 3);
        #pragma unroll
        for (int r = 0; r < 8; ++r) {
            const int gr = mbase + r;
            if (gr < nrows)
                Y[(size_t)gr * 512 + ocol] = a[r] + bv;
        }
    };
    store_tile(acc00, 0, 0);
    store_tile(acc01, 0, 1);
    store_tile(acc10, 1, 0);
    store_tile(acc11, 1, 1);
}

extern "C" void kernel_launch(void* const* d_in, const int* in_sizes, int n_in,
                              void* d_out, int out_size, void* d_ws, size_t ws_size,
                              hipStream_t stream) {
    (void)in_sizes; (void)d_ws; (void)ws_size; (void)out_size;
    // setup_inputs() dict order: x_t, W_t, b_t per type, for
    // t in {paper, author, institution, field, venue, topic}
    static const int KDIM[6] = {128, 256, 512, 384, 768, 256};
    static const int NROW[6] = {100000, 50000, 75000, 40000, 25000, 60000};

    if (n_in < 18) return;

    size_t out_off = 0;
    for (int i = 0; i < 6; ++i) {
        const float* x = (const float*)d_in[3 * i + 0];
        const float* w = (const float*)d_in[3 * i + 1];
        const float* b = (const float*)d_in[3 * i + 2];
        float* y = (float*)d_out + out_off;

        dim3 grid((NROW[i] + BM - 1) / BM, 512 / BN);
        dim3 block(256);
        hipLaunchKernelGGL(hdl_gemm_f32, grid, block, 0, stream,
                           x, w, b, y, NROW[i], KDIM[i]);
        out_off += (size_t)NROW[i] * 512;
    }
}